// GCN_32822140076791
// MI455X (gfx1250) — compile-verified
//
#include <hip/hip_runtime.h>
#include <hip/hip_bf16.h>

// GCN 2-layer: gcn_conv -> BN -> LeakyReLU -> gcn_conv -> BN -> LeakyReLU
// N=50000 nodes, E=600000 edges, feat 256 -> 128 -> 128, all fp32.

#define GCN_N     50000
#define GCN_E     600000
#define GCN_F1    256
#define GCN_F2    128
#define GCN_EPS   1e-5f
#define GCN_SLOPE 0.1f

typedef float v2f __attribute__((ext_vector_type(2)));
typedef float v8f __attribute__((ext_vector_type(8)));

// gfx1250 async global->LDS copy (ASYNCcnt-tracked); fall back to float4 staging
// if the builtins are not available in this toolchain.
// Probe-derived signature: void(v4i AS1* src, v4i AS3* dst, int offset, imm cpol)
#if defined(__gfx1250__) && \
    __has_builtin(__builtin_amdgcn_global_load_async_to_lds_b128) && \
    __has_builtin(__builtin_amdgcn_s_wait_asynccnt)
#define GCN_HAS_ASYNC 1
typedef int v4i __attribute__((vector_size(16)));
typedef __attribute__((address_space(1))) v4i* gv4p;   // global (AS1) v4i*
typedef __attribute__((address_space(3))) v4i* sv4p;   // LDS    (AS3) v4i*
#else
#define GCN_HAS_ASYNC 0
#endif

// ---------------------------------------------------------------------------
// small utility kernels
// ---------------------------------------------------------------------------
__global__ void fill_kernel(float* __restrict__ p, float v, int n) {
    int i = blockIdx.x * blockDim.x + threadIdx.x;
    if (i < n) p[i] = v;
}

__global__ void deg_count_kernel(const long long* __restrict__ col,
                                 float* __restrict__ deg, int e) {
    int i = blockIdx.x * blockDim.x + threadIdx.x;
    if (i < e) atomicAdd(&deg[(int)col[i]], 1.0f);
}

__global__ void dinv_kernel(const float* __restrict__ deg,
                            float* __restrict__ dinv, int n) {
    int i = blockIdx.x * blockDim.x + threadIdx.x;
    if (i < n) {
        float d = deg[i];                      // >= 1 always (self loop)
        dinv[i] = d > 0.0f ? rsqrtf(d) : 0.0f;
    }
}

// ---------------------------------------------------------------------------
// GEMM: C[M x 128] = A[M x K] * B[K x 128] + bias, fp32 WMMA 16x16x4.
// One block = 16 rows of A (staged in LDS, async copy); 8 waves own one
// 16x16 tile each (covers all 128 output columns).
// A-fragment (ISA 7.12.2, 32-bit A 16x4): lanes 0-15 K={0,1}, lanes 16-31 K={2,3}.
// C/D (32-bit 16x16): VGPR j -> row j (lanes 0-15) / row j+8 (lanes 16-31).
// ---------------------------------------------------------------------------
template <int K>
__global__ __launch_bounds__(256)
void gemm_wmma_f32_kernel(const float* __restrict__ A,
                          const float* __restrict__ B,     // [K x 128] row-major
                          const float* __restrict__ bias,  // [128]
                          float* __restrict__ C) {
    constexpr int LDSK = K + 4;                 // +16B pad kills LDS bank conflicts
    __shared__ __align__(16) float As[16 * LDSK];
    const int m0  = blockIdx.x * 16;
    const int tid = threadIdx.x;

    constexpr int KV   = K / 4;                 // float4s per row
    constexpr int NVEC = 16 * KV;               // float4s per 16-row stripe

#if GCN_HAS_ASYNC
    for (int i = tid; i < NVEC; i += 256) {
        int r  = i / KV;
        int cc = i - r * KV;
        __builtin_amdgcn_global_load_async_to_lds_b128(
            (gv4p)(A + (long long)(m0 + r) * K + cc * 4),
            (sv4p)(As + r * LDSK + cc * 4),
            /*offset=*/0, /*cpol=*/0);
    }
    __builtin_amdgcn_s_wait_asynccnt(0);
    __syncthreads();
#else
    for (int i = tid; i < NVEC; i += 256) {
        int r  = i / KV;
        int cc = i - r * KV;
        float4 v = ((const float4*)(A + (long long)(m0 + r) * K))[cc];
        ((float4*)(As + r * LDSK))[cc] = v;
    }
    __syncthreads();
#endif

    const int wave = tid >> 5;
    const int lane = tid & 31;
    const int half = lane >> 4;   // 0: K pair {0,1} / rows 0-7,  1: {2,3} / rows 8-15
    const int l16  = lane & 15;
    const int n0   = wave * 16;

    v8f c;
    const float bv = bias[n0 + l16];   // bias is per-column, constant across rows
#pragma unroll
    for (int j = 0; j < 8; ++j) c[j] = bv;

    const float* arow = As + l16 * LDSK;
    const float* bcol = B + n0 + l16;
#pragma unroll 8
    for (int k = 0; k < K; k += 4) {
        v2f a, b;
        a.x = arow[k + 2 * half + 0];
        a.y = arow[k + 2 * half + 1];
        b.x = bcol[(k + 2 * half + 0) * GCN_F2];
        b.y = bcol[(k + 2 * half + 1) * GCN_F2];
        c = __builtin_amdgcn_wmma_f32_16x16x4_f32(
                /*neg_a=*/false, a, /*neg_b=*/false, b,
                /*c_mod=*/(short)0, c, /*reuse_a=*/false, /*reuse_b=*/false);
    }

    float* out = C + (long long)m0 * GCN_F2 + n0 + l16;
#pragma unroll
    for (int j = 0; j < 8; ++j)
        out[(long long)(j + 8 * half) * GCN_F2] = c[j];
}

// ---------------------------------------------------------------------------
// self-loop init: out[i][:] = dinv[i]^2 * h[i][:]   (full overwrite, no atomics)
// one thread per float4 -> n = N*32 threads
// ---------------------------------------------------------------------------
__global__ void selfloop_kernel(const float* __restrict__ h,
                                const float* __restrict__ dinv,
                                float* __restrict__ out, int n) {
    int i = blockIdx.x * blockDim.x + threadIdx.x;
    if (i >= n) return;
    int   r = i >> 5;
    float d = dinv[r];
    float s = d * d;
    float4 v = ((const float4*)h)[i];
    float4 o; o.x = s * v.x; o.y = s * v.y; o.z = s * v.z; o.w = s * v.w;
    ((float4*)out)[i] = o;
}

// ---------------------------------------------------------------------------
// edge scatter: out[col] += dinv[row]*dinv[col] * h[row]
// one wave32 per edge; 128 floats = 4 per lane (one 512B row per wave)
// ---------------------------------------------------------------------------
__global__ __launch_bounds__(256)
void scatter_kernel(const float* __restrict__ h,
                    const float* __restrict__ dinv,
                    const long long* __restrict__ row,
                    const long long* __restrict__ col,
                    float* __restrict__ out, int e) {
    int wave = (blockIdx.x * blockDim.x + threadIdx.x) >> 5;
    int lane = threadIdx.x & 31;
    if (wave >= e) return;
    int r = (int)row[wave];
    int c = (int)col[wave];
    float nrm = dinv[r] * dinv[c];
    float4 v = ((const float4*)(h + (long long)r * GCN_F2))[lane];
    float* op = out + (long long)c * GCN_F2 + lane * 4;
    atomicAdd(op + 0, nrm * v.x);
    atomicAdd(op + 1, nrm * v.y);
    atomicAdd(op + 2, nrm * v.z);
    atomicAdd(op + 3, nrm * v.w);
}

// ---------------------------------------------------------------------------
// BatchNorm stats: stats[0..127]=sum, stats[128..255]=sumsq (atomic block sums)
// 256 threads: thread t handles feature t&127, rows strided 2 per block-step
// ---------------------------------------------------------------------------
__global__ __launch_bounds__(256)
void bn_stats_kernel(const float* __restrict__ h, float* __restrict__ stats) {
    __shared__ float ssum[256];
    __shared__ float ssq[256];
    const int tid = threadIdx.x;
    const int f   = tid & 127;
    float s = 0.0f, q = 0.0f;
    for (int r = blockIdx.x * 2 + (tid >> 7); r < GCN_N; r += gridDim.x * 2) {
        float v = h[(long long)r * GCN_F2 + f];
        s += v;
        q += v * v;
    }
    ssum[tid] = s;
    ssq[tid]  = q;
    __syncthreads();
    if (tid < 128) {
        atomicAdd(&stats[f],       ssum[tid] + ssum[tid + 128]);
        atomicAdd(&stats[128 + f], ssq[tid]  + ssq[tid + 128]);
    }
}

// ---------------------------------------------------------------------------
// BN apply (+ LeakyReLU), biased variance: var = E[x^2] - E[x]^2
// one thread per float4
// ---------------------------------------------------------------------------
__global__ void bn_apply_kernel(const float* __restrict__ hin,
                                float* __restrict__ out,
                                const float* __restrict__ stats,
                                const float* __restrict__ gamma,
                                const float* __restrict__ beta, int n) {
    int i = blockIdx.x * blockDim.x + threadIdx.x;
    if (i >= n) return;
    const float inv_n = 1.0f / (float)GCN_N;
    const int   f4    = (i & 31) * 4;
    float4 v = ((const float4*)hin)[i];
    float r[4] = {v.x, v.y, v.z, v.w};
#pragma unroll
    for (int j = 0; j < 4; ++j) {
        int   f   = f4 + j;
        float mu  = stats[f] * inv_n;
        float var = stats[128 + f] * inv_n - mu * mu;
        float sc  = gamma[f] * rsqrtf(var + GCN_EPS);
        float y   = (r[j] - mu) * sc + beta[f];
        r[j] = y >= 0.0f ? y : GCN_SLOPE * y;
    }
    float4 o; o.x = r[0]; o.y = r[1]; o.z = r[2]; o.w = r[3];
    ((float4*)out)[i] = o;
}

// ---------------------------------------------------------------------------
extern "C" void kernel_launch(void* const* d_in, const int* in_sizes, int n_in,
                              void* d_out, int out_size, void* d_ws, size_t ws_size,
                              hipStream_t stream) {
    const float*     x   = (const float*)d_in[0];
    const long long* ei  = (const long long*)d_in[1];
    const float*     W1  = (const float*)d_in[2];
    const float*     b1  = (const float*)d_in[3];
    const float*     W2  = (const float*)d_in[4];
    const float*     b2  = (const float*)d_in[5];
    const float*     g1  = (const float*)d_in[6];
    const float*     be1 = (const float*)d_in[7];
    const float*     g2  = (const float*)d_in[8];
    const float*     be2 = (const float*)d_in[9];
    float*           out = (float*)d_out;

    const int E = in_sizes[1] / 2;
    const long long* erow = ei;
    const long long* ecol = ei + E;

    // workspace layout (floats): deg[N] | dinv[N] | stats[256] | bufA[N*128] | bufB[N*128]
    float* ws    = (float*)d_ws;
    float* deg   = ws;
    float* dinv  = deg + GCN_N;
    float* stats = dinv + GCN_N;
    float* bufA  = stats + 256;                       // 16B-aligned offsets
    float* bufB  = bufA + (long long)GCN_N * GCN_F2;

    const int TB = 256;
    const int gN    = (GCN_N + TB - 1) / TB;
    const int gE    = (E + TB - 1) / TB;
    const int gElem = (GCN_N * 32 + TB - 1) / TB;     // one thread per float4
    const int gScat = (E * 32 + TB - 1) / TB;         // one wave per edge
    const int gGemm = GCN_N / 16;                     // 3125 stripes of 16 rows
    const int gStat = 512;

    // --- degree / normalization (shared by both layers) ---
    fill_kernel<<<gN, TB, 0, stream>>>(deg, 1.0f, GCN_N);   // self loop
    deg_count_kernel<<<gE, TB, 0, stream>>>(ecol, deg, E);
    dinv_kernel<<<gN, TB, 0, stream>>>(deg, dinv, GCN_N);

    // --- layer 1: h1 = x @ W1 + b1 ; aggregate ; BN+LeakyReLU ---
    gemm_wmma_f32_kernel<GCN_F1><<<gGemm, TB, 0, stream>>>(x, W1, b1, bufA);
    selfloop_kernel<<<gElem, TB, 0, stream>>>(bufA, dinv, bufB, GCN_N * 32);
    scatter_kernel<<<gScat, TB, 0, stream>>>(bufA, dinv, erow, ecol, bufB, E);
    fill_kernel<<<1, TB, 0, stream>>>(stats, 0.0f, 256);
    bn_stats_kernel<<<gStat, TB, 0, stream>>>(bufB, stats);
    bn_apply_kernel<<<gElem, TB, 0, stream>>>(bufB, bufB, stats, g1, be1, GCN_N * 32);

    // --- layer 2: h2 = a1 @ W2 + b2 ; aggregate into d_out ; BN+LeakyReLU ---
    gemm_wmma_f32_kernel<GCN_F2><<<gGemm, TB, 0, stream>>>(bufB, W2, b2, bufA);
    selfloop_kernel<<<gElem, TB, 0, stream>>>(bufA, dinv, out, GCN_N * 32);
    scatter_kernel<<<gScat, TB, 0, stream>>>(bufA, dinv, erow, ecol, out, E);
    fill_kernel<<<1, TB, 0, stream>>>(stats, 0.0f, 256);
    bn_stats_kernel<<<gStat, TB, 0, stream>>>(out, stats);
    bn_apply_kernel<<<gElem, TB, 0, stream>>>(out, out, stats, g2, be2, GCN_N * 32);
}